// GraphGeGLU_6880537608489
// MI455X (gfx1250) — compile-verified
//
#include <hip/hip_runtime.h>
#include <math.h>

#define DIN  128
#define DHID 256
#define DOUT 128

typedef __attribute__((ext_vector_type(2))) float v2f;
typedef __attribute__((ext_vector_type(8))) float v8f;

// ---------------- degree / normalization ----------------

__global__ __launch_bounds__(256) void k_init_deg(float* __restrict__ deg, int n) {
  int i = blockIdx.x * 256 + threadIdx.x;
  if (i < n) deg[i] = 1.0f;  // self loop contributes 1 to target-side degree
}

__global__ __launch_bounds__(256) void k_count_deg(const int* __restrict__ ei,
                                                   float* __restrict__ deg, int E) {
  int e = blockIdx.x * 256 + threadIdx.x;
  if (e < E) atomicAdd(&deg[ei[E + e]], 1.0f);  // col = edge_index[1] = target
}

__global__ __launch_bounds__(256) void k_rsqrt(float* __restrict__ deg, int n) {
  int i = blockIdx.x * 256 + threadIdx.x;
  if (i < n) {
    float d = deg[i];
    deg[i] = (d > 0.0f) ? rsqrtf(d) : 0.0f;
  }
}

// ---------------- h = x @ W via fp32 WMMA 16x16x4 ----------------
// Block: 256 threads = 8 waves. Each wave computes one 16x16 tile of h.
// grid.x = row tiles (16 rows each), grid.y = 2 (each covers 128 of 256 cols).

#define XS_PITCH 132  // 128 + 4 pad: conflict-free float2 reads, 16B-aligned float4 staging

__global__ __launch_bounds__(256) void k_gemm_wmma(const float* __restrict__ x,
                                                   const float* __restrict__ W,
                                                   float* __restrict__ h, int n) {
  __shared__ __align__(16) float xs[16 * XS_PITCH];

  const int row0 = blockIdx.x * 16;
  // stage 16x128 fp32 x-tile into LDS (512 float4s, 2 per thread)
  for (int i = threadIdx.x; i < 16 * (DIN / 4); i += 256) {
    int r  = i >> 5;   // 32 float4 per row
    int c4 = i & 31;
    int gr = row0 + r;
    if (gr > n - 1) gr = n - 1;  // clamp (n is a multiple of 16 here anyway)
    *(float4*)(xs + r * XS_PITCH + c4 * 4) =
        ((const float4*)(x + (size_t)gr * DIN))[c4];
  }
  __syncthreads();

  const int wave = threadIdx.x >> 5;
  const int lane = threadIdx.x & 31;
  const int col0 = blockIdx.y * 128 + wave * 16;
  const int m    = lane & 15;          // row within tile (A) / col within tile (B,D)
  const int kh   = (lane >> 4) << 1;   // 0 for lanes 0-15, 2 for lanes 16-31

  v8f acc = {};
  const float* wp = W + (size_t)kh * DHID + col0 + m;

#pragma unroll 8
  for (int k0 = 0; k0 < DIN; k0 += 4) {
    // A 16x4: lane holds x[m][k0+kh], x[m][k0+kh+1]
    float2 av = *(const float2*)(xs + m * XS_PITCH + k0 + kh);
    v2f a; a.x = av.x; a.y = av.y;
    // B 4x16: lane holds W[k0+kh][col0+m], W[k0+kh+1][col0+m]
    v2f b;
    b.x = wp[(size_t)(k0)     * DHID];
    b.y = wp[(size_t)(k0 + 1) * DHID];
    acc = __builtin_amdgcn_wmma_f32_16x16x4_f32(false, a, false, b,
                                                (short)0, acc, false, false);
  }

  // D layout: VGPR v -> row rbase+v, col = col0 + m.
  // Single wave-uniform guard; full-tile path is 8 plain stores off one base
  // pointer with immediate offsets (v * DHID * 4 bytes fits IOFFSET).
  const int rbase = row0 + ((lane >> 4) << 3);
  float* hp = h + (size_t)rbase * DHID + col0 + m;
  if (row0 + 16 <= n) {
#pragma unroll
    for (int v = 0; v < 8; ++v) {
      hp[(size_t)v * DHID] = acc[v];
    }
  } else {
#pragma unroll
    for (int v = 0; v < 8; ++v) {
      if (rbase + v < n) hp[(size_t)v * DHID] = acc[v];
    }
  }
}

// ---------------- agg = dinv^2 * h + b  (self loop + bias) ----------------

__global__ __launch_bounds__(256) void k_init_agg(const float* __restrict__ h,
                                                  const float* __restrict__ dinv,
                                                  const float* __restrict__ bias,
                                                  float* __restrict__ agg, int n) {
  int t = blockIdx.x * 256 + threadIdx.x;  // one float4 per thread
  int total = n * (DHID / 4);
  if (t >= total) return;
  int i  = t >> 6;   // 64 float4 per node
  int d4 = t & 63;
  float di = dinv[i];
  float w  = di * di;
  float4 hv = ((const float4*)(h + (size_t)i * DHID))[d4];
  float4 bv = ((const float4*)bias)[d4];
  float4 r;
  r.x = w * hv.x + bv.x;
  r.y = w * hv.y + bv.y;
  r.z = w * hv.z + bv.z;
  r.w = w * hv.w + bv.w;
  ((float4*)(agg + (size_t)i * DHID))[d4] = r;
}

// ---------------- edge scatter: agg[c] += dinv[r]*dinv[c]*h[r] ----------------
// One wave32 per edge; each lane handles 8 contiguous floats (2x float4 load + 8 atomics).

__global__ __launch_bounds__(256) void k_scatter(const int* __restrict__ ei,
                                                 const float* __restrict__ h,
                                                 const float* __restrict__ dinv,
                                                 float* __restrict__ agg, int E) {
  int w    = (blockIdx.x * 256 + threadIdx.x) >> 5;
  int lane = threadIdx.x & 31;
  if (w >= E) return;
  int r = ei[w];          // source
  int c = ei[E + w];      // target
  float s = dinv[r] * dinv[c];
  const float4* hp = (const float4*)(h + (size_t)r * DHID + lane * 8);
  float*        ap = agg + (size_t)c * DHID + lane * 8;
  float4 h0 = hp[0];
  float4 h1 = hp[1];
  atomicAdd(ap + 0, s * h0.x);
  atomicAdd(ap + 1, s * h0.y);
  atomicAdd(ap + 2, s * h0.z);
  atomicAdd(ap + 3, s * h0.w);
  atomicAdd(ap + 4, s * h1.x);
  atomicAdd(ap + 5, s * h1.y);
  atomicAdd(ap + 6, s * h1.z);
  atomicAdd(ap + 7, s * h1.w);
}

// ---------------- out = a * gelu_exact(gate) ----------------

__device__ __forceinline__ float gelu_exact(float g) {
  return 0.5f * g * (1.0f + erff(g * 0.70710678118654752440f));
}

__global__ __launch_bounds__(256) void k_finalize(const float* __restrict__ agg,
                                                  float* __restrict__ out, int n) {
  int t = blockIdx.x * 256 + threadIdx.x;  // one float4 of output
  int total = n * (DOUT / 4);
  if (t >= total) return;
  int i  = t >> 5;   // 32 float4 per node
  int j4 = t & 31;
  float4 a = ((const float4*)(agg + (size_t)i * DHID))[j4];
  float4 g = ((const float4*)(agg + (size_t)i * DHID + DOUT))[j4];
  float4 o;
  o.x = a.x * gelu_exact(g.x);
  o.y = a.y * gelu_exact(g.y);
  o.z = a.z * gelu_exact(g.z);
  o.w = a.w * gelu_exact(g.w);
  ((float4*)(out + (size_t)i * DOUT))[j4] = o;
}

// ---------------- launcher ----------------

extern "C" void kernel_launch(void* const* d_in, const int* in_sizes, int n_in,
                              void* d_out, int out_size, void* d_ws, size_t ws_size,
                              hipStream_t stream) {
  const float* x    = (const float*)d_in[0];
  const int*   ei   = (const int*)d_in[1];   // [2, E] int
  const float* W    = (const float*)d_in[2]; // [128, 256]
  const float* bias = (const float*)d_in[3]; // [256]
  float*       out  = (float*)d_out;         // [N, 128]

  const int n = in_sizes[0] / DIN;
  const int E = in_sizes[1] / 2;

  size_t hbytes = (((size_t)n * DHID * sizeof(float)) + 255) & ~(size_t)255;
  char* ws   = (char*)d_ws;
  float* h    = (float*)(ws);                 // [n, 256]
  float* agg  = (float*)(ws + hbytes);        // [n, 256]
  float* dinv = (float*)(ws + 2 * hbytes);    // [n] (deg -> rsqrt(deg) in place)

  k_init_deg <<<(n + 255) / 256, 256, 0, stream>>>(dinv, n);
  k_count_deg<<<(E + 255) / 256, 256, 0, stream>>>(ei, dinv, E);
  k_rsqrt    <<<(n + 255) / 256, 256, 0, stream>>>(dinv, n);

  dim3 ggrid((n + 15) / 16, 2);
  k_gemm_wmma<<<ggrid, 256, 0, stream>>>(x, W, h, n);

  int t_agg = n * (DHID / 4);
  k_init_agg<<<(t_agg + 255) / 256, 256, 0, stream>>>(h, dinv, bias, agg, n);

  long long t_sc = (long long)E * 32;
  k_scatter<<<(unsigned)((t_sc + 255) / 256), 256, 0, stream>>>(ei, h, dinv, agg, E);

  int t_fin = n * (DOUT / 4);
  k_finalize<<<(t_fin + 255) / 256, 256, 0, stream>>>(agg, out, n);
}